// CausalLocalSelfFlexBlock_34084860461625
// MI455X (gfx1250) — compile-verified
//
#include <hip/hip_runtime.h>
#include <math.h>

// MI455X / gfx1250, wave32. All matmuls use V_WMMA_F32_16X16X4_F32 (f32 in, f32 acc)
// since the layer is HBM-bound on this chip (23.3 TB/s); f32 WMMA throughput far
// exceeds the bandwidth roofline and keeps numerics at reference precision.

typedef __attribute__((ext_vector_type(2))) float v2f;
typedef __attribute__((ext_vector_type(8))) float v8f;

#define D_MODEL 512
#define HEADS   8
#define HDIM    64
#define SEQ     8192
#define NTOK    32768   // B*L = 4*8192
#define DFF     2048
#define WIN     512

__device__ __forceinline__ v8f v8zero() {
  v8f z;
#pragma unroll
  for (int i = 0; i < 8; ++i) z[i] = 0.f;
  return z;
}

__device__ __forceinline__ v8f wmma4(v2f a, v2f b, v8f c) {
  // D = A(16x4,f32) * B(4x16,f32) + C(16x16,f32)
  return __builtin_amdgcn_wmma_f32_16x16x4_f32(false, a, false, b, (short)0, c,
                                               false, false);
}

// ---------------------------------------------------------------------------
// LayerNorm: one wave (32 lanes) per 512-dim row, 8 rows per 256-thread block.
// ---------------------------------------------------------------------------
__global__ __launch_bounds__(256) void ln_kernel(const float* __restrict__ x,
                                                 const float* __restrict__ w,
                                                 const float* __restrict__ b,
                                                 float* __restrict__ out) {
  const int lane = threadIdx.x & 31;
  const int row  = blockIdx.x * 8 + (threadIdx.x >> 5);
  const float* xr = x + (size_t)row * D_MODEL;
  float v[16];
#pragma unroll
  for (int c = 0; c < 4; ++c) {
    float4 t = *(const float4*)(xr + (size_t)(c * 32 + lane) * 4);
    v[c * 4 + 0] = t.x; v[c * 4 + 1] = t.y; v[c * 4 + 2] = t.z; v[c * 4 + 3] = t.w;
  }
  float s = 0.f;
#pragma unroll
  for (int i = 0; i < 16; ++i) s += v[i];
#pragma unroll
  for (int off = 16; off >= 1; off >>= 1) s += __shfl_xor(s, off, 32);
  const float mu = s * (1.f / 512.f);
  float vs = 0.f;
#pragma unroll
  for (int i = 0; i < 16; ++i) { float d = v[i] - mu; vs += d * d; }
#pragma unroll
  for (int off = 16; off >= 1; off >>= 1) vs += __shfl_xor(vs, off, 32);
  const float rs = rsqrtf(vs * (1.f / 512.f) + 1e-5f);
  float* orow = out + (size_t)row * D_MODEL;
#pragma unroll
  for (int c = 0; c < 4; ++c) {
    const int c0 = (c * 32 + lane) * 4;
    float4 wv = *(const float4*)(w + c0);
    float4 bv = *(const float4*)(b + c0);
    float4 o;
    o.x = (v[c * 4 + 0] - mu) * rs * wv.x + bv.x;
    o.y = (v[c * 4 + 1] - mu) * rs * wv.y + bv.y;
    o.z = (v[c * 4 + 2] - mu) * rs * wv.z + bv.z;
    o.w = (v[c * 4 + 3] - mu) * rs * wv.w + bv.w;
    *(float4*)(orow + c0) = o;
  }
}

// ---------------------------------------------------------------------------
// WMMA f32 GEMM tile: 128 threads (4 waves), block tile 64x64, each wave a
// 32x32 subtile (2x2 of 16x16 frags). LDS-staged A(64x16) and B(16x64).
// ---------------------------------------------------------------------------
__device__ __forceinline__ void gemm_tile(const float* __restrict__ A, int lda,
                                          const float* __restrict__ B, int ldb,
                                          int K, int m0, int n0, float* ldsA,
                                          float* ldsB, v8f c[2][2]) {
  const int tid  = threadIdx.x;
  const int lane = tid & 31;
  const int wv   = tid >> 5;
  const int wm   = (wv >> 1) << 5;
  const int wn   = (wv & 1) << 5;
  const int lm   = lane & 15;
  const int ksel = (lane >> 4) << 1;  // 0 or 2 (K split across lane halves)
  c[0][0] = v8zero(); c[0][1] = v8zero(); c[1][0] = v8zero(); c[1][1] = v8zero();
  for (int k0 = 0; k0 < K; k0 += 16) {
    __syncthreads();
#pragma unroll
    for (int t = 0; t < 2; ++t) {
      const int idx = tid + t * 128;
      const int ra = idx >> 2, qa = idx & 3;
      *(float4*)(ldsA + ra * 16 + qa * 4) =
          *(const float4*)(A + (size_t)(m0 + ra) * lda + k0 + qa * 4);
      const int rb = idx >> 4, qb = idx & 15;
      *(float4*)(ldsB + rb * 64 + qb * 4) =
          *(const float4*)(B + (size_t)(k0 + rb) * ldb + n0 + qb * 4);
    }
    __syncthreads();
#pragma unroll
    for (int kk = 0; kk < 4; ++kk) {
      v2f a[2], bf[2];
#pragma unroll
      for (int i = 0; i < 2; ++i) {
        const float* ap = ldsA + (wm + i * 16 + lm) * 16 + kk * 4 + ksel;
        a[i].x = ap[0]; a[i].y = ap[1];
      }
#pragma unroll
      for (int j = 0; j < 2; ++j) {
        const float* bp = ldsB + (kk * 4 + ksel) * 64 + wn + j * 16 + lm;
        bf[j].x = bp[0]; bf[j].y = bp[64];
      }
#pragma unroll
      for (int i = 0; i < 2; ++i)
#pragma unroll
        for (int j = 0; j < 2; ++j) c[i][j] = wmma4(a[i], bf[j], c[i][j]);
    }
  }
}

// Dual-B variant for the gated FFN (computes a-tile and gate-tile together).
__device__ __forceinline__ void gemm_tile2(const float* __restrict__ A, int lda,
                                           const float* __restrict__ B1,
                                           const float* __restrict__ B2, int ldb,
                                           int K, int m0, int n0, float* ldsA,
                                           float* ldsB1, float* ldsB2,
                                           v8f c1[2][2], v8f c2[2][2]) {
  const int tid  = threadIdx.x;
  const int lane = tid & 31;
  const int wv   = tid >> 5;
  const int wm   = (wv >> 1) << 5;
  const int wn   = (wv & 1) << 5;
  const int lm   = lane & 15;
  const int ksel = (lane >> 4) << 1;
  c1[0][0] = v8zero(); c1[0][1] = v8zero(); c1[1][0] = v8zero(); c1[1][1] = v8zero();
  c2[0][0] = v8zero(); c2[0][1] = v8zero(); c2[1][0] = v8zero(); c2[1][1] = v8zero();
  for (int k0 = 0; k0 < K; k0 += 16) {
    __syncthreads();
#pragma unroll
    for (int t = 0; t < 2; ++t) {
      const int idx = tid + t * 128;
      const int ra = idx >> 2, qa = idx & 3;
      *(float4*)(ldsA + ra * 16 + qa * 4) =
          *(const float4*)(A + (size_t)(m0 + ra) * lda + k0 + qa * 4);
      const int rb = idx >> 4, qb = idx & 15;
      *(float4*)(ldsB1 + rb * 64 + qb * 4) =
          *(const float4*)(B1 + (size_t)(k0 + rb) * ldb + n0 + qb * 4);
      *(float4*)(ldsB2 + rb * 64 + qb * 4) =
          *(const float4*)(B2 + (size_t)(k0 + rb) * ldb + n0 + qb * 4);
    }
    __syncthreads();
#pragma unroll
    for (int kk = 0; kk < 4; ++kk) {
      v2f a[2], b1f[2], b2f[2];
#pragma unroll
      for (int i = 0; i < 2; ++i) {
        const float* ap = ldsA + (wm + i * 16 + lm) * 16 + kk * 4 + ksel;
        a[i].x = ap[0]; a[i].y = ap[1];
      }
#pragma unroll
      for (int j = 0; j < 2; ++j) {
        const float* bp1 = ldsB1 + (kk * 4 + ksel) * 64 + wn + j * 16 + lm;
        b1f[j].x = bp1[0]; b1f[j].y = bp1[64];
        const float* bp2 = ldsB2 + (kk * 4 + ksel) * 64 + wn + j * 16 + lm;
        b2f[j].x = bp2[0]; b2f[j].y = bp2[64];
      }
#pragma unroll
      for (int i = 0; i < 2; ++i)
#pragma unroll
        for (int j = 0; j < 2; ++j) {
          c1[i][j] = wmma4(a[i], b1f[j], c1[i][j]);
          c2[i][j] = wmma4(a[i], b2f[j], c2[i][j]);
        }
    }
  }
}

// ---------------------------------------------------------------------------
// QKV GEMM: h1[32768,512] @ Wqkv[512,1536] + bqkv, scatter to q/k/v [B,H,L,64]
// ---------------------------------------------------------------------------
__global__ __launch_bounds__(128) void gemm_qkv_kernel(
    const float* __restrict__ A, const float* __restrict__ Bw,
    const float* __restrict__ bias, float* __restrict__ qb,
    float* __restrict__ kb, float* __restrict__ vb) {
  __shared__ float ldsA[64 * 16];
  __shared__ float ldsB[16 * 64];
  v8f c[2][2];
  const int m0 = blockIdx.y << 6, n0 = blockIdx.x << 6;
  gemm_tile(A, D_MODEL, Bw, 3 * D_MODEL, D_MODEL, m0, n0, ldsA, ldsB, c);
  const int lane = threadIdx.x & 31, wv = threadIdx.x >> 5;
  const int wm = (wv >> 1) << 5, wn = (wv & 1) << 5;
  const int lm = lane & 15, hi = lane >> 4;
#pragma unroll
  for (int i = 0; i < 2; ++i)
#pragma unroll
    for (int j = 0; j < 2; ++j)
#pragma unroll
      for (int r = 0; r < 8; ++r) {
        const int row = m0 + wm + i * 16 + r + hi * 8;
        const int col = n0 + wn + j * 16 + lm;
        const float val = c[i][j][r] + bias[col];
        const int part = col >> 9, c2 = col & 511;
        const int head = c2 >> 6, dh = c2 & 63;
        const int bb = row >> 13, l = row & (SEQ - 1);
        float* dst = part == 0 ? qb : (part == 1 ? kb : vb);
        dst[(((size_t)bb * HEADS + head) * SEQ + l) * HDIM + dh] = val;
      }
}

// ---------------------------------------------------------------------------
// RoPE in place over a [B,H,L,64] buffer; one thread per rotation pair.
// ---------------------------------------------------------------------------
__global__ __launch_bounds__(256) void rope_kernel(float* __restrict__ buf) {
  const size_t idx = (size_t)blockIdx.x * 256 + threadIdx.x;  // B*H*L*32
  const int i = (int)(idx & 31);
  const int l = (int)((idx >> 5) & (SEQ - 1));
  const size_t bh = idx >> 18;  // 8192*32 = 2^18
  const float inv = powf(10000.f, -(float)i / 32.f);
  float sn, cs;
  sincosf((float)l * inv, &sn, &cs);
  const size_t base = (bh * SEQ + (size_t)l) << 6;
  const float x1 = buf[base + i];
  const float x2 = buf[base + i + 32];
  buf[base + i]      = x1 * cs - x2 * sn;
  buf[base + i + 32] = x2 * cs + x1 * sn;
}

// ---------------------------------------------------------------------------
// Flash-style 512-window causal attention. Block = 2 waves = 64 query rows of
// one (b,h). Each wave owns 32 rows x 64 keys per tile; online softmax with
// width-16 shuffle row reductions (C-layout rows live in 16-lane halves).
// ---------------------------------------------------------------------------
__global__ __launch_bounds__(64) void attn_kernel(
    const float* __restrict__ qb, const float* __restrict__ kb,
    const float* __restrict__ vb, const unsigned char* __restrict__ pad,
    float* __restrict__ attn) {
  __shared__ float Qs[64 * 64];
  __shared__ float Ks[64 * 64];
  __shared__ float Vs[64 * 64];
  __shared__ float Ps[64 * 64];
  __shared__ float padS[64];

  const int qt = blockIdx.x, h = blockIdx.y, bB = blockIdx.z;
  const int tid = threadIdx.x;
  const int lane = tid & 31, w = tid >> 5;
  const int lm = lane & 15, hi = lane >> 4;
  const int ksel = hi << 1;
  const int q0 = qt << 6;
  const int wq = w << 5;  // this wave's query-row offset in the 64-row tile
  const size_t headBase = ((size_t)(bB * HEADS + h)) * SEQ * HDIM;

  // Load Q tile (64x64)
#pragma unroll
  for (int c = 0; c < 16; ++c) {
    const int idx = c * 64 + tid;  // float4 index
    const int r = idx >> 4, q4 = idx & 15;
    *(float4*)(Qs + r * 64 + q4 * 4) =
        *(const float4*)(qb + headBase + (size_t)(q0 + r) * HDIM + q4 * 4);
  }

  v8f o[2][4];
#pragma unroll
  for (int i = 0; i < 2; ++i)
#pragma unroll
    for (int j = 0; j < 4; ++j) o[i][j] = v8zero();
  float mrow[2][8], lrow[2][8];
#pragma unroll
  for (int i = 0; i < 2; ++i)
#pragma unroll
    for (int r = 0; r < 8; ++r) { mrow[i][r] = -1e30f; lrow[i][r] = 0.f; }

  const int kt0 = (q0 - WIN > 0 ? q0 - WIN : 0) >> 6;
  const int kt1 = q0 >> 6;
  for (int kt = kt0; kt <= kt1; ++kt) {
    const int kbase = kt << 6;
    __syncthreads();
#pragma unroll
    for (int c = 0; c < 16; ++c) {
      const int idx = c * 64 + tid;
      const int r = idx >> 4, q4 = idx & 15;
      *(float4*)(Ks + r * 64 + q4 * 4) =
          *(const float4*)(kb + headBase + (size_t)(kbase + r) * HDIM + q4 * 4);
      *(float4*)(Vs + r * 64 + q4 * 4) =
          *(const float4*)(vb + headBase + (size_t)(kbase + r) * HDIM + q4 * 4);
    }
    padS[tid] = pad[(size_t)bB * SEQ + kbase + tid] ? 1.f : 0.f;
    __syncthreads();

    // S = Q @ K^T (wave: 32 rows x 64 keys)
    v8f s[2][4];
#pragma unroll
    for (int i = 0; i < 2; ++i)
#pragma unroll
      for (int j = 0; j < 4; ++j) s[i][j] = v8zero();
    for (int k4 = 0; k4 < 64; k4 += 4) {
      v2f a[2], bf[4];
#pragma unroll
      for (int i = 0; i < 2; ++i) {
        const float* ap = Qs + (wq + i * 16 + lm) * 64 + k4 + ksel;
        a[i].x = ap[0]; a[i].y = ap[1];
      }
#pragma unroll
      for (int j = 0; j < 4; ++j) {
        const float* bp = Ks + (j * 16 + lm) * 64 + k4 + ksel;  // B[k][n]=K[n][k]
        bf[j].x = bp[0]; bf[j].y = bp[1];
      }
#pragma unroll
      for (int i = 0; i < 2; ++i)
#pragma unroll
        for (int j = 0; j < 4; ++j) s[i][j] = wmma4(a[i], bf[j], s[i][j]);
    }

    // Mask + online softmax update
#pragma unroll
    for (int i = 0; i < 2; ++i)
#pragma unroll
      for (int r = 0; r < 8; ++r) {
        const int qg = q0 + wq + i * 16 + r + hi * 8;
        float rmax = -1e30f;
#pragma unroll
        for (int j = 0; j < 4; ++j) {
          const int kg = kbase + j * 16 + lm;
          float sv = s[i][j][r] * 0.125f;  // HDIM^-0.5
          const bool ok = (kg <= qg) && (qg - kg <= WIN) && (padS[j * 16 + lm] == 0.f);
          sv = ok ? sv : -1e30f;
          s[i][j][r] = sv;
          rmax = fmaxf(rmax, sv);
        }
        rmax = fmaxf(rmax, __shfl_xor(rmax, 1, 16));
        rmax = fmaxf(rmax, __shfl_xor(rmax, 2, 16));
        rmax = fmaxf(rmax, __shfl_xor(rmax, 4, 16));
        rmax = fmaxf(rmax, __shfl_xor(rmax, 8, 16));
        const float mnew = fmaxf(mrow[i][r], rmax);
        const float alpha = expf(mrow[i][r] - mnew);
        float rsum = 0.f;
#pragma unroll
        for (int j = 0; j < 4; ++j) {
          const float p = expf(s[i][j][r] - mnew);
          s[i][j][r] = p;
          rsum += p;
        }
        rsum += __shfl_xor(rsum, 1, 16);
        rsum += __shfl_xor(rsum, 2, 16);
        rsum += __shfl_xor(rsum, 4, 16);
        rsum += __shfl_xor(rsum, 8, 16);
        lrow[i][r] = lrow[i][r] * alpha + rsum;
        mrow[i][r] = mnew;
#pragma unroll
        for (int j = 0; j < 4; ++j) o[i][j][r] *= alpha;
      }

    // Stage P through LDS to re-fragment as a WMMA A operand
#pragma unroll
    for (int i = 0; i < 2; ++i)
#pragma unroll
      for (int j = 0; j < 4; ++j)
#pragma unroll
        for (int r = 0; r < 8; ++r)
          Ps[(wq + i * 16 + r + hi * 8) * 64 + j * 16 + lm] = s[i][j][r];
    __syncthreads();

    // O += P @ V
    for (int k4 = 0; k4 < 64; k4 += 4) {
      v2f a[2], bf[4];
#pragma unroll
      for (int i = 0; i < 2; ++i) {
        const float* ap = Ps + (wq + i * 16 + lm) * 64 + k4 + ksel;
        a[i].x = ap[0]; a[i].y = ap[1];
      }
#pragma unroll
      for (int j = 0; j < 4; ++j) {
        const float* bp = Vs + (k4 + ksel) * 64 + j * 16 + lm;
        bf[j].x = bp[0]; bf[j].y = bp[64];
      }
#pragma unroll
      for (int i = 0; i < 2; ++i)
#pragma unroll
        for (int j = 0; j < 4; ++j) o[i][j] = wmma4(a[i], bf[j], o[i][j]);
    }
  }

  // Normalize and write to attn[B, L, H*64]
#pragma unroll
  for (int i = 0; i < 2; ++i)
#pragma unroll
    for (int r = 0; r < 8; ++r) {
      const float lr = lrow[i][r];
      const float inv = lr > 0.f ? 1.f / lr : 0.f;
      const int row = wq + i * 16 + r + hi * 8;
#pragma unroll
      for (int j = 0; j < 4; ++j) {
        const int dh = j * 16 + lm;
        attn[((size_t)bB * SEQ + q0 + row) * D_MODEL + h * HDIM + dh] =
            o[i][j][r] * inv;
      }
    }
}

// ---------------------------------------------------------------------------
// Output projection + residual: out = x + attn @ Wproj + bproj
// ---------------------------------------------------------------------------
__global__ __launch_bounds__(128) void gemm_proj_kernel(
    const float* __restrict__ A, const float* __restrict__ Bw,
    const float* __restrict__ bias, const float* __restrict__ xres,
    float* __restrict__ out) {
  __shared__ float ldsA[64 * 16];
  __shared__ float ldsB[16 * 64];
  v8f c[2][2];
  const int m0 = blockIdx.y << 6, n0 = blockIdx.x << 6;
  gemm_tile(A, D_MODEL, Bw, D_MODEL, D_MODEL, m0, n0, ldsA, ldsB, c);
  const int lane = threadIdx.x & 31, wv = threadIdx.x >> 5;
  const int wm = (wv >> 1) << 5, wn = (wv & 1) << 5;
  const int lm = lane & 15, hi = lane >> 4;
#pragma unroll
  for (int i = 0; i < 2; ++i)
#pragma unroll
    for (int j = 0; j < 2; ++j)
#pragma unroll
      for (int r = 0; r < 8; ++r) {
        const int row = m0 + wm + i * 16 + r + hi * 8;
        const int col = n0 + wn + j * 16 + lm;
        const size_t rc = (size_t)row * D_MODEL + col;
        out[rc] = xres[rc] + c[i][j][r] + bias[col];
      }
}

// ---------------------------------------------------------------------------
// FFN GEMM1 (gated): gact = silu(h2@W1[:, :2048]+b1a) * (h2@W1[:, 2048:]+b1g)
// ---------------------------------------------------------------------------
__global__ __launch_bounds__(128) void gemm_glu_kernel(
    const float* __restrict__ A, const float* __restrict__ W1,
    const float* __restrict__ b1, float* __restrict__ gact) {
  __shared__ float ldsA[64 * 16];
  __shared__ float ldsB1[16 * 64];
  __shared__ float ldsB2[16 * 64];
  v8f c1[2][2], c2[2][2];
  const int m0 = blockIdx.y << 6, n0 = blockIdx.x << 6;
  gemm_tile2(A, D_MODEL, W1, W1 + DFF, 2 * DFF, D_MODEL, m0, n0, ldsA, ldsB1,
             ldsB2, c1, c2);
  const int lane = threadIdx.x & 31, wv = threadIdx.x >> 5;
  const int wm = (wv >> 1) << 5, wn = (wv & 1) << 5;
  const int lm = lane & 15, hi = lane >> 4;
#pragma unroll
  for (int i = 0; i < 2; ++i)
#pragma unroll
    for (int j = 0; j < 2; ++j)
#pragma unroll
      for (int r = 0; r < 8; ++r) {
        const int row = m0 + wm + i * 16 + r + hi * 8;
        const int col = n0 + wn + j * 16 + lm;
        const float a = c1[i][j][r] + b1[col];
        const float g = c2[i][j][r] + b1[col + DFF];
        gact[(size_t)row * DFF + col] = (a / (1.f + expf(-a))) * g;
      }
}

// ---------------------------------------------------------------------------
// FFN GEMM2 + residual accumulate into d_out (which holds x2)
// ---------------------------------------------------------------------------
__global__ __launch_bounds__(128) void gemm2_kernel(const float* __restrict__ A,
                                                    const float* __restrict__ Bw,
                                                    const float* __restrict__ bias,
                                                    float* __restrict__ out) {
  __shared__ float ldsA[64 * 16];
  __shared__ float ldsB[16 * 64];
  v8f c[2][2];
  const int m0 = blockIdx.y << 6, n0 = blockIdx.x << 6;
  gemm_tile(A, DFF, Bw, D_MODEL, DFF, m0, n0, ldsA, ldsB, c);
  const int lane = threadIdx.x & 31, wv = threadIdx.x >> 5;
  const int wm = (wv >> 1) << 5, wn = (wv & 1) << 5;
  const int lm = lane & 15, hi = lane >> 4;
#pragma unroll
  for (int i = 0; i < 2; ++i)
#pragma unroll
    for (int j = 0; j < 2; ++j)
#pragma unroll
      for (int r = 0; r < 8; ++r) {
        const int row = m0 + wm + i * 16 + r + hi * 8;
        const int col = n0 + wn + j * 16 + lm;
        const size_t rc = (size_t)row * D_MODEL + col;
        out[rc] = out[rc] + c[i][j][r] + bias[col];
      }
}

// ---------------------------------------------------------------------------
extern "C" void kernel_launch(void* const* d_in, const int* in_sizes, int n_in,
                              void* d_out, int out_size, void* d_ws,
                              size_t ws_size, hipStream_t stream) {
  const float* x            = (const float*)d_in[0];
  const unsigned char* pad  = (const unsigned char*)d_in[1];  // bool mask
  const float* ln1w = (const float*)d_in[2];
  const float* ln1b = (const float*)d_in[3];
  const float* Wqkv = (const float*)d_in[4];
  const float* bqkv = (const float*)d_in[5];
  const float* Wpro = (const float*)d_in[6];
  const float* bpro = (const float*)d_in[7];
  const float* ln2w = (const float*)d_in[8];
  const float* ln2b = (const float*)d_in[9];
  const float* W1   = (const float*)d_in[10];
  const float* b1   = (const float*)d_in[11];
  const float* W2   = (const float*)d_in[12];
  const float* b2   = (const float*)d_in[13];
  float* out = (float*)d_out;

  // Workspace layout (floats). Total = NTOK*(512*4 + 2048) = 512 MB.
  float* ws = (float*)d_ws;
  const size_t NT = NTOK;
  float* h1   = ws;                  // NT*512  (reused as attn output)
  float* qb   = h1 + NT * 512;       // NT*512  (reused as h2 after attention)
  float* kb   = qb + NT * 512;       // NT*512
  float* vb   = kb + NT * 512;       // NT*512
  float* gact = vb + NT * 512;       // NT*2048
  float* attnb = h1;
  float* h2    = qb;

  ln_kernel<<<NTOK / 8, 256, 0, stream>>>(x, ln1w, ln1b, h1);
  gemm_qkv_kernel<<<dim3(24, NTOK / 64), 128, 0, stream>>>(h1, Wqkv, bqkv, qb, kb, vb);
  rope_kernel<<<32768, 256, 0, stream>>>(qb);
  rope_kernel<<<32768, 256, 0, stream>>>(kb);
  attn_kernel<<<dim3(SEQ / 64, HEADS, 4), 64, 0, stream>>>(qb, kb, vb, pad, attnb);
  gemm_proj_kernel<<<dim3(8, NTOK / 64), 128, 0, stream>>>(attnb, Wpro, bpro, x, out);
  ln_kernel<<<NTOK / 8, 256, 0, stream>>>(out, ln2w, ln2b, h2);
  gemm_glu_kernel<<<dim3(32, NTOK / 64), 128, 0, stream>>>(h2, W1, b1, gact);
  gemm2_kernel<<<dim3(8, NTOK / 64), 128, 0, stream>>>(gact, W2, b2, out);
}